// ChannelSymmetry_42339787603992
// MI455X (gfx1250) — compile-verified
//
#include <hip/hip_runtime.h>

// out[b,c,t] = X[b, src, t],  src = swap_mask[b][c>>1] ? (c^1) : c
// Pure permutation copy: 0 FLOPs, ~524 MB HBM traffic -> bandwidth bound
// (floor ~22.5 us at 23.3 TB/s). Strategy: b128 vector VMEM with non-temporal
// cache policy both directions (footprint 2.6x the 192 MB L2, zero reuse),
// one 8-wave block per channel row, coalesced along T, 4x unrolled main loop
// so each wave keeps 4 global_load_b128 in flight (LOADcnt pipelining).

typedef __attribute__((ext_vector_type(4))) float f32x4;

#define CS_C    64       // channels
#define CS_LOGC 6
#define CS_NT   256      // block size (8 wave32)

__global__ __launch_bounds__(CS_NT) void ChannelSymmetry_copy_kernel(
    const float* __restrict__ X,
    const int* __restrict__ swap_mask,   // [B, 16] int32 (bool promoted)
    float* __restrict__ out,
    int T, int T4)
{
    const int row = blockIdx.x;              // row = b*C + c
    const int b   = row >> CS_LOGC;
    const int c   = row & (CS_C - 1);
    const int p   = c >> 1;                  // pair index 0..31

    const int m   = swap_mask[b * 16 + p];
    const int src = m ? (c ^ 1) : c;

    const size_t in_base  = ((size_t)b * CS_C + (size_t)src) * (size_t)T;
    const size_t out_base = (size_t)row * (size_t)T;

    const f32x4* __restrict__ in4  = (const f32x4*)(X + in_base);
    f32x4* __restrict__       out4 = (f32x4*)(out + out_base);

    const int stride = CS_NT;
    int i = threadIdx.x;

    // Main streaming loop, 4x unrolled: 4 NT b128 loads in flight, then
    // 4 NT b128 stores. 2 KB per wave per iteration.
    for (; i + 3 * stride < T4; i += 4 * stride) {
        f32x4 v0 = __builtin_nontemporal_load(in4 + i);
        f32x4 v1 = __builtin_nontemporal_load(in4 + i + stride);
        f32x4 v2 = __builtin_nontemporal_load(in4 + i + 2 * stride);
        f32x4 v3 = __builtin_nontemporal_load(in4 + i + 3 * stride);
        __builtin_nontemporal_store(v0, out4 + i);
        __builtin_nontemporal_store(v1, out4 + i + stride);
        __builtin_nontemporal_store(v2, out4 + i + 2 * stride);
        __builtin_nontemporal_store(v3, out4 + i + 3 * stride);
    }

    // Leftover full vectors.
    for (; i < T4; i += stride) {
        f32x4 v = __builtin_nontemporal_load(in4 + i);
        __builtin_nontemporal_store(v, out4 + i);
    }

    // Scalar tail if T % 4 != 0 (not hit for T=8000, kept for generality).
    for (int t = (T4 << 2) + threadIdx.x; t < T; t += CS_NT) {
        float v = __builtin_nontemporal_load(X + in_base + t);
        __builtin_nontemporal_store(v, out + out_base + t);
    }
}

extern "C" void kernel_launch(void* const* d_in, const int* in_sizes, int n_in,
                              void* d_out, int out_size, void* d_ws, size_t ws_size,
                              hipStream_t stream) {
    (void)n_in; (void)out_size; (void)d_ws; (void)ws_size;

    const float* X         = (const float*)d_in[0];
    const int*   swap_mask = (const int*)d_in[1];   // [B,16] bool -> int32
    float*       out       = (float*)d_out;

    const int P = 16;
    const int B = in_sizes[1] / P;                 // 128
    const int T = in_sizes[0] / (B * CS_C);        // 8000
    const int T4 = T >> 2;

    dim3 grid(B * CS_C);    // 8192 row-blocks
    dim3 block(CS_NT);      // 8 waves (wave32)
    ChannelSymmetry_copy_kernel<<<grid, block, 0, stream>>>(X, swap_mask, out, T, T4);
}